// SamplingClassifier_15547781611623
// MI455X (gfx1250) — compile-verified
//
#include <hip/hip_runtime.h>

#define N_NODES 100000
#define N_EDGES 3200000
#define N_REL   16
#define EMB     32
#define N_CLS   50
#define BATCH   1024
#define N_SEG   (N_REL * N_NODES)

typedef __attribute__((ext_vector_type(2))) float v2f;
typedef __attribute__((ext_vector_type(8))) float v8f;

// ---- degree: one f32 atomic per edge --------------------------------------
__global__ void deg_kernel(const int* __restrict__ src, const int* __restrict__ rel,
                           float* __restrict__ deg) {
  int e = blockIdx.x * blockDim.x + threadIdx.x;
  if (e >= N_EDGES) return;
  int seg = rel[e] * N_NODES + src[e];
  atomicAdd(&deg[seg], 1.0f);
}

// ---- invert degrees in place ----------------------------------------------
__global__ void inv_kernel(float* __restrict__ deg) {
  int i = blockIdx.x * blockDim.x + threadIdx.x;
  if (i >= N_SEG) return;
  float d = deg[i];
  deg[i] = d > 0.0f ? 1.0f / d : 0.0f;
}

// ---- Ht[r][n][i] = sum_j h[n][j] * W[r][i][j]  (optionally relu(h)) -------
// One wave per 16-node tile. A fragments (h tile, K=32) loaded ONCE and
// reused across all 16 rels: 16 rels x (8 K-steps x 2 col-tiles) = 256
// chained V_WMMA_F32_16X16X4_F32 per wave, f32 end to end.
__global__ void __launch_bounds__(256)
gemm_ht_kernel(const float* __restrict__ h, const float* __restrict__ W,
               float* __restrict__ Ht, int apply_relu) {
  const int lane = threadIdx.x & 31;
  const int wave = threadIdx.x >> 5;
  const int tile = blockIdx.x * 8 + wave;
  if (tile * 16 >= N_NODES) return;            // wave-uniform: EXEC all-1s inside
  const int nb = tile * 16;

  // A fragment (16x4 f32): lanes 0-15 -> M=lane, K={0,1}; lanes 16-31 -> K={2,3}
  const int m    = lane & 15;
  const int koff = (lane >> 4) << 1;

  v2f a[8];
  const float* ap = h + (size_t)(nb + m) * EMB + koff;
#pragma unroll
  for (int jc = 0; jc < 8; ++jc) a[jc] = *(const v2f*)(ap + jc * 4);
  if (apply_relu) {
#pragma unroll
    for (int jc = 0; jc < 8; ++jc) {
      a[jc].x = a[jc].x > 0.0f ? a[jc].x : 0.0f;
      a[jc].y = a[jc].y > 0.0f ? a[jc].y : 0.0f;
    }
  }

  // D layout: VGPR v holds rows M = v + 8*(lane>>4), N = lane&15
  const int mrow = (lane >> 4) << 3;
  const int ncol = lane & 15;

  for (int r = 0; r < N_REL; ++r) {
    // B(k, i) = W[r][ib + i][j0 + k]
    const float* bp0 = W + ((size_t)r * EMB + m) * EMB + koff;
    const float* bp1 = bp0 + 16 * EMB;
    v8f c0 = {}, c1 = {};
#pragma unroll
    for (int jc = 0; jc < 8; ++jc) {
      v2f b0 = *(const v2f*)(bp0 + jc * 4);
      v2f b1 = *(const v2f*)(bp1 + jc * 4);
      c0 = __builtin_amdgcn_wmma_f32_16x16x4_f32(false, a[jc], false, b0,
                                                 (short)0, c0, false, false);
      c1 = __builtin_amdgcn_wmma_f32_16x16x4_f32(false, a[jc], false, b1,
                                                 (short)0, c1, false, false);
    }
    float* orow = Ht + ((size_t)r * N_NODES + nb + mrow) * EMB;
#pragma unroll
    for (int v = 0; v < 8; ++v) {
      orow[(size_t)v * EMB + ncol]      = c0[v];
      orow[(size_t)v * EMB + 16 + ncol] = c1[v];
    }
  }
}

// ---- edge scatter: hpre[src] += inv_deg[(rel,src)] * Ht[rel][dst] ---------
// One edge per wave32: 128B coalesced gather + 128B atomic row into the
// 12.8MB (L2-resident) accumulator.
__global__ void edge_scatter(const int* __restrict__ src, const int* __restrict__ rel,
                             const int* __restrict__ dst, const float* __restrict__ inv_deg,
                             const float* __restrict__ Ht, float* __restrict__ hpre) {
  int lane = threadIdx.x & 31;
  int e = blockIdx.x * (blockDim.x >> 5) + (threadIdx.x >> 5);
  if (e >= N_EDGES) return;
  int s = src[e], r = rel[e], d = dst[e];
  float val = inv_deg[r * N_NODES + s];
  float t = Ht[((size_t)r * N_NODES + d) * EMB + lane];
  atomicAdd(&hpre[(size_t)s * EMB + lane], val * t);
}

// ---- classifier: out[n,c] = relu(h[n,:]) . cls_w[c,:] + cls_b[c] ---------
__global__ void cls_kernel(const float* __restrict__ h, const float* __restrict__ cw,
                           const float* __restrict__ cb, float* __restrict__ out) {
  int idx = blockIdx.x * blockDim.x + threadIdx.x;
  if (idx >= BATCH * N_CLS) return;
  int n = idx / N_CLS, c = idx % N_CLS;
  float acc = cb[c];
  const float* hr = h + (size_t)n * EMB;
  const float* wr = cw + (size_t)c * EMB;
#pragma unroll
  for (int j = 0; j < EMB; ++j) {
    float hv = hr[j]; hv = hv > 0.0f ? hv : 0.0f;
    acc += hv * wr[j];
  }
  out[idx] = acc;
}

extern "C" void kernel_launch(void* const* d_in, const int* in_sizes, int n_in,
                              void* d_out, int out_size, void* d_ws, size_t ws_size,
                              hipStream_t stream) {
  (void)in_sizes; (void)n_in; (void)out_size; (void)ws_size;
  const float* emb0  = (const float*)d_in[0];
  const float* W1    = (const float*)d_in[1];
  const float* W2    = (const float*)d_in[2];
  const float* cls_w = (const float*)d_in[3];
  const float* cls_b = (const float*)d_in[4];
  const int*   src   = (const int*)d_in[5];
  const int*   rel   = (const int*)d_in[6];
  const int*   dst   = (const int*)d_in[7];
  float* out = (float*)d_out;

  // workspace (floats): Ht[N_SEG*EMB] | deg[N_SEG] | hpre1[N*EMB] | hpre2[N*EMB]
  float* Ht    = (float*)d_ws;
  float* deg   = Ht + (size_t)N_SEG * EMB;
  float* hpre1 = deg + (size_t)N_SEG;
  float* hpre2 = hpre1 + (size_t)N_NODES * EMB;

  hipMemsetAsync(deg, 0, (size_t)N_SEG * sizeof(float), stream);
  hipMemsetAsync(hpre1, 0, (size_t)N_NODES * EMB * sizeof(float), stream);
  hipMemsetAsync(hpre2, 0, (size_t)N_NODES * EMB * sizeof(float), stream);

  deg_kernel<<<(N_EDGES + 255) / 256, 256, 0, stream>>>(src, rel, deg);
  inv_kernel<<<(N_SEG + 255) / 256, 256, 0, stream>>>(deg);

  const int gemm_grid = (N_NODES / 16 + 7) / 8;

  // layer 1: transform-then-scatter (Ht fully overwritten, no zeroing needed)
  gemm_ht_kernel<<<gemm_grid, 256, 0, stream>>>(emb0, W1, Ht, 0);
  edge_scatter<<<(N_EDGES + 7) / 8, 256, 0, stream>>>(src, rel, dst, deg, Ht, hpre1);

  // layer 2: relu of layer-1 result fused into the GEMM's A loads
  gemm_ht_kernel<<<gemm_grid, 256, 0, stream>>>(hpre1, W2, Ht, 1);
  edge_scatter<<<(N_EDGES + 7) / 8, 256, 0, stream>>>(src, rel, dst, deg, Ht, hpre2);

  // classifier (final relu fused into its reads)
  cls_kernel<<<(BATCH * N_CLS + 255) / 256, 256, 0, stream>>>(hpre2, cls_w, cls_b, out);
}